// AttentionConv_80762565034317
// MI455X (gfx1250) — compile-verified
//
#include <hip/hip_runtime.h>
#include <math.h>

// ---------------------------------------------------------------------------
// AttentionConv (B=2, C=64, H=W=32) for gfx1250 (MI455X), fp32 WMMA path
// + async global->LDS staging of V (ASYNCcnt) and bank-conflict-free LDS.
//
// Masked-key folding: score at any masked upsampled position m is
//   s_mask(n) = -1e8 * sum_d q_up[d,n]   (identical for all 3072 masked m)
// and v_up is zero there, so masked columns only enter softmax max/denominator:
//   m_tot  = max(max_unmasked, s_mask)
//   denom  = sum_unmasked exp(s - m_tot) + 3072 * exp(s_mask - m_tot)
// Exactly matches the reference math while shrinking scores to 4096 x 1024.
// ---------------------------------------------------------------------------

typedef __attribute__((ext_vector_type(2))) float v2f;
typedef __attribute__((ext_vector_type(8))) float v8f;

#define CH   64
#define DQ   8
#define NP   1024     // pixels per image (32*32) == unmasked key count
#define NUP  4096     // upsampled positions (64*64)
#define VST  1028     // padded row stride (floats): +16B so the 16 rows start
                      // 4 banks apart -> b64 PV reads tile all 64 LDS banks

// -------------------------- kernel 1: q/k/v 1x1 convs ----------------------
__global__ __launch_bounds__(256) void qkv_kernel(
    const float* __restrict__ x,
    const float* __restrict__ Wq, const float* __restrict__ Wk,
    const float* __restrict__ Wv,
    float* __restrict__ qout, float* __restrict__ kout, float* __restrict__ vout)
{
    __shared__ float wts[80 * 64];          // rows 0..7 Wq, 8..15 Wk, 16..79 Wv
    for (int i = threadIdx.x; i < 80 * 64; i += 256) {
        int row = i >> 6, col = i & 63;
        float w;
        if (row < 8)        w = Wq[row * 64 + col];
        else if (row < 16)  w = Wk[(row - 8) * 64 + col];
        else                w = Wv[(row - 16) * 64 + col];
        wts[i] = w;
    }
    __syncthreads();

    int gid = blockIdx.x * 256 + threadIdx.x;   // 0..2047 : (batch, pixel)
    int b = gid >> 10, p = gid & 1023;
    const float* xp = x + (size_t)b * CH * NP + p;

    float xr[64];
    #pragma unroll
    for (int c = 0; c < 64; ++c) xr[c] = xp[c * NP];

    #pragma unroll
    for (int o = 0; o < 8; ++o) {
        float aq = 0.f, ak = 0.f;
        #pragma unroll
        for (int c = 0; c < 64; ++c) {
            aq = fmaf(wts[o * 64 + c],        xr[c], aq);
            ak = fmaf(wts[(o + 8) * 64 + c],  xr[c], ak);
        }
        qout[b * DQ * NP + o * NP + p] = aq;
        kout[b * DQ * NP + o * NP + p] = ak;
    }
    for (int o = 0; o < 64; ++o) {
        float av = 0.f;
        #pragma unroll
        for (int c = 0; c < 64; ++c)
            av = fmaf(wts[(o + 16) * 64 + c], xr[c], av);
        vout[b * CH * NP + o * NP + p] = av;
    }
}

// ---- async stage one v chunk (16 rows x 1024 floats) into padded LDS ------
// 64KB as 4096 x 16B pieces; 32 lanes x B128 -> 128 async instructions.
__device__ __forceinline__ void issue_v_chunk(const float* gsrc,
                                              unsigned lds_base_b, int lane)
{
    unsigned long long g = (unsigned long long)gsrc;
    #pragma unroll 1
    for (int it = 0; it < 128; ++it) {
        int e = it * 32 + lane;                 // 16B-piece index
        int r = e >> 8;                         // row 0..15
        int c = e & 255;                        // 16B unit within row
        unsigned voff = (unsigned)(r * 4096 + c * 16);          // global bytes
        unsigned loff = lds_base_b + (unsigned)(r * (VST * 4) + c * 16);
        asm volatile("global_load_async_to_lds_b128 %0, %1, %2"
                     :: "v"(loff), "v"(voff), "s"(g) : "memory");
    }
}

// ------------------- kernel 2: fused attention (1 wave / 16 rows) ----------
__global__ __launch_bounds__(32) void attn_kernel(
    const float* __restrict__ q, const float* __restrict__ k,
    const float* __restrict__ v, float* __restrict__ out)
{
    __shared__ float sc[16 * VST];     // score strip -> P=exp(s-m), padded
    __shared__ float vbuf[16 * VST];   // async-staged v chunk, padded
    __shared__ float srow[16];         // sum_d q_up per row
    __shared__ float minv[16];         // 1/denom per row

    const int  lane = threadIdx.x;
    const int  b    = blockIdx.y;
    const int  tile = blockIdx.x;           // 0..255, 16 query rows each
    const int  row  = lane & 15;
    const bool hi   = lane >= 16;

    const float* qb = q + (size_t)b * DQ * NP;
    const float* kb = k + (size_t)b * DQ * NP;
    const float* vb = v + (size_t)b * CH * NP;

    unsigned vbase = (unsigned)(unsigned long long)
        (__attribute__((address_space(3))) char*)(void*)vbuf;

    // kick off async staging of v chunk 0; overlaps QK WMMAs + softmax
    issue_v_chunk(vb, vbase, lane);

    // ---- q_up (bilinear, align_corners) for this lane's row & 4 channels ---
    // fp32 16x16x4 A layout: lanes 0-15 hold K=0,1 ; lanes 16-31 hold K=2,3.
    int   n  = tile * 16 + row;
    int   yy = n >> 6, xx = n & 63;
    float py = (float)yy * (31.0f / 63.0f);
    float px = (float)xx * (31.0f / 63.0f);
    int   iy = (int)py; if (iy > 30) iy = 30;
    int   ix = (int)px; if (ix > 30) ix = 30;
    float wy = py - (float)iy, wx = px - (float)ix;
    float w00 = (1.f - wy) * (1.f - wx), w01 = (1.f - wy) * wx;
    float w10 = wy * (1.f - wx),         w11 = wy * wx;
    int   p00 = iy * 32 + ix;
    int   c0  = hi ? 2 : 0;

    float qc[4];
    float s4 = 0.f;
    #pragma unroll
    for (int j = 0; j < 4; ++j) {
        int d = c0 + (j & 1) + (j >> 1) * 4;        // {c0, c0+1, c0+4, c0+5}
        const float* qd = qb + d * NP;
        float val = w00 * qd[p00] + w01 * qd[p00 + 1]
                  + w10 * qd[p00 + 32] + w11 * qd[p00 + 33];
        qc[j] = val; s4 += val;
    }
    v2f a0; a0.x = qc[0]; a0.y = qc[1];             // K = c0, c0+1
    v2f a1; a1.x = qc[2]; a1.y = qc[3];             // K = c0+4, c0+5
    float s8 = s4 + __shfl_xor(s4, 16, 32);         // all 8 channels
    if (!hi) srow[row] = s8;

    // ---------------- phase 1: scores = q_up^T k over 1024 keys ------------
    for (int kt = 0; kt < 64; ++kt) {
        int p = kt * 16 + row;                      // key pixel index
        v2f b0; b0.x = kb[(c0 + 0) * NP + p]; b0.y = kb[(c0 + 1) * NP + p];
        v2f b1; b1.x = kb[(c0 + 4) * NP + p]; b1.y = kb[(c0 + 5) * NP + p];
        v8f s = {};
        s = __builtin_amdgcn_wmma_f32_16x16x4_f32(false, a0, false, b0,
                                                  (short)0, s, false, false);
        s = __builtin_amdgcn_wmma_f32_16x16x4_f32(false, a1, false, b1,
                                                  (short)0, s, false, false);
        int col   = kt * 16 + row;                  // N = lane&15
        int rbase = hi ? 8 : 0;                     // M = vgpr (+8 hi half)
        #pragma unroll
        for (int g = 0; g < 8; ++g) sc[(rbase + g) * VST + col] = s[g];
    }
    __syncthreads();

    // ---------------- phase 2: softmax stats (masked cols analytic) --------
    {
        int j0 = hi ? 512 : 0;
        const float* rd = &sc[row * VST + j0];
        float mx = -3.0e38f;
        for (int j = 0; j < 512; ++j) mx = fmaxf(mx, rd[j]);
        mx = fmaxf(mx, __shfl_xor(mx, 16, 32));
        float smask = -1e8f * srow[row];
        float m = fmaxf(mx, smask);

        float* wr = &sc[row * VST + j0];
        float psum = 0.f;
        for (int j = 0; j < 512; ++j) {
            float e = __expf(wr[j] - m);
            wr[j] = e; psum += e;
        }
        psum += __shfl_xor(psum, 16, 32);
        float denom = psum + 3072.0f * __expf(smask - m);
        if (!hi) minv[row] = 1.0f / denom;
    }
    __syncthreads();

    // ---------------- phase 3: out = (P @ V^T) / denom via WMMA ------------
    // B operand streamed from the async-staged, padded LDS chunk.
    for (int cc = 0; cc < 4; ++cc) {
        asm volatile("s_wait_asynccnt 0x0" ::: "memory");   // chunk cc landed

        v8f acc = {};
        const float* prow = &sc[row * VST + (hi ? 2 : 0)];
        const float* vrow = &vbuf[row * VST + (hi ? 2 : 0)];
        for (int ks = 0; ks < 256; ++ks) {
            v2f pa = *(const v2f*)(prow + ks * 4);  // A: P[row][k..k+1 (+2 hi)]
            v2f vr = *(const v2f*)(vrow + ks * 4);  // B: v[ch][k..k+1 (+2 hi)]
            acc = __builtin_amdgcn_wmma_f32_16x16x4_f32(false, pa, false, vr,
                                                        (short)0, acc,
                                                        false, false);
        }

        if (cc < 3) {
            // all LDS reads of vbuf done before async engine overwrites (WAR)
            asm volatile("s_wait_dscnt 0x0" ::: "memory");
            issue_v_chunk(vb + (cc + 1) * 16 * NP, vbase, lane);
        }

        int ch    = cc * 16 + row;                  // N = lane&15
        int rbase = hi ? 8 : 0;
        float* op = out + ((size_t)b * CH + ch) * NUP + tile * 16 + rbase;
        #pragma unroll
        for (int g = 0; g < 8; ++g) op[g] = acc[g] * minv[rbase + g];
    }
}

// --------------------------------- launch ----------------------------------
extern "C" void kernel_launch(void* const* d_in, const int* in_sizes, int n_in,
                              void* d_out, int out_size, void* d_ws, size_t ws_size,
                              hipStream_t stream)
{
    const float* x  = (const float*)d_in[0];
    const float* Wq = (const float*)d_in[1];
    const float* Wk = (const float*)d_in[2];
    const float* Wv = (const float*)d_in[3];

    float* ws = (float*)d_ws;                  // 640 KB used
    float* qv = ws;                            // 2*8*1024
    float* kv = ws + 2 * DQ * NP;              // 2*8*1024
    float* vv = ws + 4 * DQ * NP;              // 2*64*1024

    qkv_kernel<<<dim3(8), dim3(256), 0, stream>>>(x, Wq, Wk, Wv, qv, kv, vv);
    attn_kernel<<<dim3(256, 2), dim3(32), 0, stream>>>(qv, kv, vv, (float*)d_out);
}